// NDDE_1D_90228672954672
// MI455X (gfx1250) — compile-verified
//
#include <hip/hip_runtime.h>

// ---------------------------------------------------------------------------
// NDDE forward-Euler solver, persistent cooperative kernel for gfx1250.
//   x' = W2 @ tanh(Wx@x + Wy@x(t-tau) + b1) + b2,  dt = tau/10, 20000 steps.
//
// MI455X reasoning: all weights (24MB fp32) live in the 192MB L2; per-step cost
// is sequential L2 streaming + 2 grid barriers. The delayed term Wy@y is
// batched over the 10-step delay window into an fp32 WMMA GEMM
// (v_wmma_f32_16x16x4_f32, exact precision vs the fp32 reference). x/h vectors
// are broadcast-staged into LDS via GLOBAL_LOAD_ASYNC_TO_LDS_B128 so vector
// traffic doesn't double the weight traffic. Trajectory stores are
// non-temporal to keep weights resident in L2.
// ---------------------------------------------------------------------------

#define NTAU    10
#define D_DIM   1024
#define H_DIM   2048
#define NBLOCKS 256
#define TPB     256   // 8 wave32 per block -> 2048 waves total == H_DIM rows

typedef float v2f __attribute__((ext_vector_type(2)));
typedef float v8f __attribute__((ext_vector_type(8)));
typedef int   v4i __attribute__((ext_vector_type(4)));

#if defined(__has_builtin)
#  if __has_builtin(__builtin_amdgcn_global_load_async_to_lds_b128) && \
      __has_builtin(__builtin_amdgcn_s_wait_asynccnt)
#    define USE_ASYNC_LDS 1
#  endif
#endif
#ifndef USE_ASYNC_LDS
#  define USE_ASYNC_LDS 0
#endif

__device__ __forceinline__ float wave_reduce_sum(float v) {
#pragma unroll
  for (int off = 16; off > 0; off >>= 1)
    v += __shfl_xor(v, off, 32);
  return v;
}

// Stage a global vector into LDS, one b128 per thread per iter, then sync.
__device__ __forceinline__ void lds_stage(float* dst, const float* src, int n) {
  const int tid = threadIdx.x;
#if USE_ASYNC_LDS
  for (int i = tid * 4; i < n; i += TPB * 4)
    __builtin_amdgcn_global_load_async_to_lds_b128(
        (__attribute__((address_space(1))) v4i*)(src + i),
        (__attribute__((address_space(3))) v4i*)(dst + i),
        /*offset=*/0, /*cpol=*/0);
  __builtin_amdgcn_s_wait_asynccnt(0);
#else
  for (int i = tid * 4; i < n; i += TPB * 4)
    *(float4*)(dst + i) = *(const float4*)(src + i);
#endif
  __syncthreads();
}

// Device-scope sense-reversing grid barrier. bar[0]=arrival count, bar[1]=generation.
__device__ __forceinline__ void gsync(unsigned* bar, unsigned* pgen, unsigned nblocks) {
  __threadfence();            // make this thread's stores device-visible
  __syncthreads();
  if (threadIdx.x == 0) {
    const unsigned g = *pgen;
    const unsigned prev =
        __hip_atomic_fetch_add(&bar[0], 1u, __ATOMIC_ACQ_REL, __HIP_MEMORY_SCOPE_AGENT);
    if (prev == nblocks - 1u) {
      __hip_atomic_store(&bar[0], 0u, __ATOMIC_RELAXED, __HIP_MEMORY_SCOPE_AGENT);
      __hip_atomic_store(&bar[1], g + 1u, __ATOMIC_RELEASE, __HIP_MEMORY_SCOPE_AGENT);
    } else {
      while (__hip_atomic_load(&bar[1], __ATOMIC_ACQUIRE, __HIP_MEMORY_SCOPE_AGENT) == g)
        __builtin_amdgcn_s_sleep(1);
    }
  }
  *pgen += 1u;
  __syncthreads();
}

// One 16x16 tile of  Cbuf = Wy @ Ypad + b1  per block, 8 waves K-splitting,
// partial 16x16 fp32 accumulators reduced through LDS.
// A layout (ISA 7.12.2, 32-bit A 16x4): lane L holds M=L&15; VGPR0/1 = K=2*(L>>4)+{0,1}.
// B layout (4x16): lane L holds N=L&15; K = 2*(L>>4)+{0,1}.
// C/D layout: lane L holds rows M = 8*(L>>4)+r (r=0..7) of column N=L&15.
__device__ __forceinline__ void wy_gemm_block(int tile,
                                              const float* __restrict__ Wy,
                                              const float* __restrict__ ring,
                                              const float* __restrict__ b1,
                                              float* __restrict__ cbuf,
                                              int wave, int lane,
                                              float* part /* LDS [8][256] */) {
  const int m0    = tile * 16;
  const int half  = lane >> 4;   // 0 or 1
  const int l15   = lane & 15;
  const int khalf = half * 2;
  const int k0    = wave * (D_DIM / 8);   // 128-wide K slice per wave

  v8f acc = {};                   // bias added after the LDS reduction
  const float* arow = Wy   + (size_t)(m0 + l15) * D_DIM + khalf;  // A row M=l15
  const float* bcol = ring + (size_t)l15 * D_DIM + khalf;         // B col N=l15

#pragma unroll 4
  for (int k = k0; k < k0 + D_DIM / 8; k += 4) {
    v2f a, b;
    a[0] = arow[k]; a[1] = arow[k + 1];
    b[0] = bcol[k]; b[1] = bcol[k + 1];
    acc = __builtin_amdgcn_wmma_f32_16x16x4_f32(
        /*neg_a=*/false, a, /*neg_b=*/false, b,
        /*c_mod=*/(short)0, acc, /*reuse_a=*/false, /*reuse_b=*/false);
  }

  // partials -> LDS: part[wave][lane*8 + r]
  float* p = part + wave * 256 + lane * 8;
#pragma unroll
  for (int r = 0; r < 8; ++r) p[r] = acc[r];
  __syncthreads();

  // 256 threads reduce 8 partials each: tid -> (L = tid>>3, r = tid&7)
  const int tid = threadIdx.x;
  const int L = tid >> 3, r = tid & 7;
  float s = 0.f;
#pragma unroll
  for (int w = 0; w < 8; ++w) s += part[w * 256 + L * 8 + r];
  const int n = L & 15;                    // column of the 16-wide (10 real) batch
  const int m = ((L >> 4) << 3) + r;       // row within the tile
  if (n < NTAU)
    cbuf[(size_t)n * H_DIM + m0 + m] = s + b1[m0 + m];
}

__global__ void __launch_bounds__(TPB)
ndde_persistent(const float* __restrict__ x0,  const float* __restrict__ tau,
                const float* __restrict__ Wx,  const float* __restrict__ Wy,
                const float* __restrict__ b1,  const float* __restrict__ W2,
                const float* __restrict__ b2,  const int* __restrict__ nsp,
                float* __restrict__ traj, unsigned* __restrict__ bar,
                float* __restrict__ ws) {
  __shared__ float xs[D_DIM];       // staged x_j          (4 KB)
  __shared__ float hs[H_DIM];       // staged tanh vector  (8 KB)
  __shared__ float part[8 * 256];   // WMMA partial tiles  (8 KB)
  __shared__ float p2sum[8];        // phase-2 wave-pair partials

  const int n_steps = nsp[0];
  const float dt    = tau[0] / (float)NTAU;
  const size_t tcols = (size_t)n_steps + 1;

  float* xbuf = ws;                       // [1024]   current state x_j
  float* ring = xbuf + D_DIM;             // [10*1024] delay ring: slot n = x_{j0-10+n}
  float* hbuf = ring + NTAU * D_DIM;      // [2048]   tanh hidden vector
  float* cbuf = hbuf + H_DIM;             // [16*2048] Wy@Y + b1, column-major

  const int tid     = threadIdx.x;
  const int lane    = tid & 31;
  const int wave    = tid >> 5;
  const int gthread = blockIdx.x * TPB + tid;
  const int gwave   = gthread >> 5;       // 0..2047
  const unsigned nb = gridDim.x;
  unsigned gen = 0;

  // ---- init: x state, trajectory column 0, delay history = x0 ----
  if (gthread < D_DIM) {
    const float v = x0[gthread];
    xbuf[gthread] = v;
    __builtin_nontemporal_store(v, traj + (size_t)gthread * tcols);
#pragma unroll
    for (int s = 0; s < NTAU; ++s) ring[s * D_DIM + gthread] = v;
  }
  gsync(bar, &gen, nb);

  for (int j = 0; j < n_steps; ++j) {
    const int col = j % NTAU;

    // ---- every 10 steps: batch the delayed term as a WMMA fp32 GEMM,
    //      one 16x16 tile per block, 8 waves K-splitting (128 blocks busy) ----
    if (col == 0) {
      if (blockIdx.x < H_DIM / 16)
        wy_gemm_block(blockIdx.x, Wy, ring, b1, cbuf, wave, lane, part);
      gsync(bar, &gen, nb);
    }

    // ---- phase 1: h = tanh(Wx @ x + cbuf[:,col]); one wave per row,
    //      x broadcast-staged into LDS once per block ----
    lds_stage(xs, xbuf, D_DIM);
    {
      const int row = gwave;
      const float* wrow = Wx + (size_t)row * D_DIM;
      float acc = 0.f;
#pragma unroll
      for (int t = 0; t < D_DIM / 128; ++t) {
        const int k = t * 128 + lane * 4;
        const float4 w  = *(const float4*)(wrow + k);
        const float4 xv = *(const float4*)(xs + k);
        acc += w.x * xv.x + w.y * xv.y + w.z * xv.z + w.w * xv.w;
      }
      acc = wave_reduce_sum(acc);
      if (lane == 0)
        hbuf[row] = tanhf(acc + cbuf[(size_t)col * H_DIM + row]);
    }
    gsync(bar, &gen, nb);

    // ---- phase 2: x_new = x + dt*(W2 @ h + b2); two waves per row
    //      (K split 1024+1024) so ALL 256 blocks pull W2 traffic ----
    lds_stage(hs, hbuf, H_DIM);
    {
      const int row   = gwave >> 1;            // 0..1023
      const int kbase = (gwave & 1) * (H_DIM / 2);
      const float* wrow = W2 + (size_t)row * H_DIM + kbase;
      const float* hsrc = hs + kbase;
      float acc = 0.f;
#pragma unroll
      for (int t = 0; t < (H_DIM / 2) / 128; ++t) {
        const int k = t * 128 + lane * 4;
        const float4 w  = *(const float4*)(wrow + k);
        const float4 hv = *(const float4*)(hsrc + k);
        acc += w.x * hv.x + w.y * hv.y + w.z * hv.z + w.w * hv.w;
      }
      acc = wave_reduce_sum(acc);
      if (lane == 0) p2sum[wave] = acc;
      __syncthreads();
      if ((wave & 1) == 0 && lane == 0) {
        const float tot = p2sum[wave] + p2sum[wave + 1];
        const float xo  = xbuf[row];
        const float xn  = fmaf(dt, tot + b2[row], xo);
        ring[(size_t)col * D_DIM + row] = xo;   // store x_j for step j+10
        xbuf[row] = xn;
        __builtin_nontemporal_store(xn, traj + (size_t)row * tcols + (size_t)(j + 1));
      }
    }
    gsync(bar, &gen, nb);
  }
}

extern "C" void kernel_launch(void* const* d_in, const int* in_sizes, int n_in,
                              void* d_out, int out_size, void* d_ws, size_t ws_size,
                              hipStream_t stream) {
  (void)in_sizes; (void)n_in; (void)out_size; (void)ws_size;
  const float* x0  = (const float*)d_in[0];
  const float* tau = (const float*)d_in[1];
  const float* Wx  = (const float*)d_in[2];
  const float* Wy  = (const float*)d_in[3];
  const float* b1  = (const float*)d_in[4];
  const float* W2  = (const float*)d_in[5];
  const float* b2  = (const float*)d_in[6];
  const int*   nsp = (const int*)d_in[7];
  float* traj = (float*)d_out;

  // Zero the grid-barrier state every call (graph-capturable async memset).
  (void)hipMemsetAsync(d_ws, 0, 256, stream);

  unsigned* bar = (unsigned*)d_ws;
  float*    ws  = (float*)d_ws + 64;   // barrier region = first 64 dwords

  ndde_persistent<<<NBLOCKS, TPB, 0, stream>>>(x0, tau, Wx, Wy, b1, W2, b2,
                                               nsp, traj, bar, ws);
}